// TTLayer_29274497089894
// MI455X (gfx1250) — compile-verified
//
#include <hip/hip_runtime.h>

typedef __attribute__((ext_vector_type(2))) float v2f;
typedef __attribute__((ext_vector_type(8))) float v8f;

#define PST 160  // paired-weight LDS row stride (dwords): kp+1 row lands +32 banks

__device__ __forceinline__ v8f wmma4_f32(v2f a, v2f b, v8f c) {
    // D = A(16x4) x B(4x16) + C, fp32 throughout
    return __builtin_amdgcn_wmma_f32_16x16x4_f32(false, a, false, b, (short)0, c, false, false);
}

// ---------------------------------------------------------------------------
// Step d=6: V6[m*64 + row*8 + r] = sum_c x[m*8+c] * core6[row*64 + c*8 + r]
// GEMM M=262144, K=8, N=64. Weights K-pair-interleaved in LDS.
// ---------------------------------------------------------------------------
__global__ __launch_bounds__(256) void tt_first(const float* __restrict__ x,
                                                const float* __restrict__ core6,
                                                float* __restrict__ out) {
    __shared__ float W[4 * PST];
    const int tid = threadIdx.x;
    for (int e = tid; e < 512; e += 256) {
        int k = e >> 6, n = e & 63;            // k = c6, n = row*8 + r
        int row = n >> 3, r = n & 7;
        W[(k >> 1) * PST + 2 * n + (k & 1)] = core6[row * 64 + k * 8 + r];
    }
    __syncthreads();

    const int lane = tid & 31;
    const int wave = tid >> 5;
    const int Lm = lane & 15;
    const int hi = lane >> 4;
    const int mbase = (blockIdx.x * 8 + wave) * 16;
    const int m = mbase + Lm;

    v8f acc0 = {}, acc1 = {}, acc2 = {}, acc3 = {};
#pragma unroll
    for (int kk = 0; kk < 2; ++kk) {
        int k0 = kk * 4 + hi * 2;
        v2f a = *(const v2f*)(x + (long)m * 8 + k0);
        const float* wr = &W[(2 * kk + hi) * PST + 2 * Lm];
        v2f b0 = *(const v2f*)(wr + 0);
        v2f b1 = *(const v2f*)(wr + 32);
        v2f b2 = *(const v2f*)(wr + 64);
        v2f b3 = *(const v2f*)(wr + 96);
        acc0 = wmma4_f32(a, b0, acc0);
        acc1 = wmma4_f32(a, b1, acc1);
        acc2 = wmma4_f32(a, b2, acc2);
        acc3 = wmma4_f32(a, b3, acc3);
    }

    float* ob = out + (long)(mbase + hi * 8) * 64 + Lm;
#pragma unroll
    for (int j = 0; j < 8; ++j) {
        ob[(long)j * 64 + 0]  = acc0[j];
        ob[(long)j * 64 + 16] = acc1[j];
        ob[(long)j * 64 + 32] = acc2[j];
        ob[(long)j * 64 + 48] = acc3[j];
    }
}

// ---------------------------------------------------------------------------
// Steps d=5..1: batched GEMM M=262144, K=64, N=64.
//   in offset for batch m, k=(c*8+r1): (m>>LB)<<(LB+6) | c<<(LB+3) | (m&BM)<<3 | r1
//   W[k][n=row*8+r] = core[row*512 + c*64 + r*8 + r1], stored K-pair-interleaved
//   out[m*64 + n].  Each wave: M=32 (two A tiles share every B fragment).
// ---------------------------------------------------------------------------
__global__ __launch_bounds__(256) void tt_mid(const float* __restrict__ in,
                                              const float* __restrict__ core,
                                              float* __restrict__ out, int LB) {
    __shared__ float W[32 * PST];
    const int tid = threadIdx.x;
    for (int e = tid; e < 4096; e += 256) {
        int k = e >> 6, n = e & 63;
        int c = k >> 3, r1 = k & 7;
        int row = n >> 3, r = n & 7;
        W[(k >> 1) * PST + 2 * n + (k & 1)] = core[row * 512 + c * 64 + r * 8 + r1];
    }
    __syncthreads();

    const int lane = tid & 31;
    const int wave = tid >> 5;
    const int Lm = lane & 15;
    const int hi = lane >> 4;
    const int mbase = (blockIdx.x * 8 + wave) * 32;
    const int m0 = mbase + Lm;
    const int m1 = mbase + 16 + Lm;
    const int msk = (1 << LB) - 1;
    const long base0 = ((long)(m0 >> LB) << (LB + 6)) + ((long)(m0 & msk) << 3);
    const long base1 = ((long)(m1 >> LB) << (LB + 6)) + ((long)(m1 & msk) << 3);

    v8f a0t0 = {}, a0t1 = {}, a0t2 = {}, a0t3 = {};
    v8f a1t0 = {}, a1t1 = {}, a1t2 = {}, a1t3 = {};
#pragma unroll
    for (int kk = 0; kk < 16; ++kk) {
        int k0 = kk * 4 + hi * 2;
        int c = k0 >> 3, r1 = k0 & 7;          // r1 even -> (r1, r1+1) contiguous
        long koff = ((long)c << (LB + 3)) + r1;
        v2f a0 = *(const v2f*)(in + base0 + koff);
        v2f a1 = *(const v2f*)(in + base1 + koff);
        const float* wr = &W[(2 * kk + hi) * PST + 2 * Lm];
        v2f b0 = *(const v2f*)(wr + 0);
        v2f b1 = *(const v2f*)(wr + 32);
        v2f b2 = *(const v2f*)(wr + 64);
        v2f b3 = *(const v2f*)(wr + 96);
        a0t0 = wmma4_f32(a0, b0, a0t0);
        a1t0 = wmma4_f32(a1, b0, a1t0);
        a0t1 = wmma4_f32(a0, b1, a0t1);
        a1t1 = wmma4_f32(a1, b1, a1t1);
        a0t2 = wmma4_f32(a0, b2, a0t2);
        a1t2 = wmma4_f32(a1, b2, a1t2);
        a0t3 = wmma4_f32(a0, b3, a0t3);
        a1t3 = wmma4_f32(a1, b3, a1t3);
    }

    float* ob0 = out + (long)(mbase + hi * 8) * 64 + Lm;
    float* ob1 = out + (long)(mbase + 16 + hi * 8) * 64 + Lm;
#pragma unroll
    for (int j = 0; j < 8; ++j) {
        ob0[(long)j * 64 + 0]  = a0t0[j];
        ob0[(long)j * 64 + 16] = a0t1[j];
        ob0[(long)j * 64 + 32] = a0t2[j];
        ob0[(long)j * 64 + 48] = a0t3[j];
        ob1[(long)j * 64 + 0]  = a1t0[j];
        ob1[(long)j * 64 + 16] = a1t1[j];
        ob1[(long)j * 64 + 32] = a1t2[j];
        ob1[(long)j * 64 + 48] = a1t3[j];
    }
}

// ---------------------------------------------------------------------------
// Step d=0: M=262144, K=64, N=8 (padded to 16). Input addressing = mid with
// LB=18 (batchA==0). W[k][n=row0] = core0[n*64 + k], cols 8..15 zero-padded,
// K-pair-interleaved (row stride 32 dwords is naturally bank-disjoint).
// Output scattered with base-8 digit reversal (the reference transpose).
// ---------------------------------------------------------------------------
__global__ __launch_bounds__(256) void tt_last(const float* __restrict__ in,
                                               const float* __restrict__ core0,
                                               float* __restrict__ out) {
    __shared__ float W[32 * 32];
    const int tid = threadIdx.x;
    for (int e = tid; e < 1024; e += 256) {
        int k = e >> 4, n = e & 15;
        W[(k >> 1) * 32 + 2 * n + (k & 1)] = (n < 8) ? core0[n * 64 + k] : 0.0f;
    }
    __syncthreads();

    const int lane = tid & 31;
    const int wave = tid >> 5;
    const int Lm = lane & 15;
    const int hi = lane >> 4;
    const int mbase = (blockIdx.x * 8 + wave) * 16;
    const int m = mbase + Lm;
    const long inBase = (long)m << 3;          // batchA = m >> 18 == 0

    v8f acc = {};
#pragma unroll
    for (int kk = 0; kk < 16; ++kk) {
        int k0 = kk * 4 + hi * 2;
        int c = k0 >> 3, r1 = k0 & 7;
        v2f a = *(const v2f*)(in + inBase + ((long)c << 21) + r1);
        v2f b = *(const v2f*)(&W[(2 * kk + hi) * 32 + 2 * Lm]);
        acc = wmma4_f32(a, b, acc);
    }

    if (Lm < 8) {
#pragma unroll
        for (int j = 0; j < 8; ++j) {
            unsigned v = (unsigned)(mbase + hi * 8 + j) * 8u + (unsigned)Lm;
            unsigned rv = 0;
#pragma unroll
            for (int i = 0; i < 7; ++i) { rv = (rv << 3) | (v & 7u); v >>= 3; }
            out[rv] = acc[j];
        }
    }
}

extern "C" void kernel_launch(void* const* d_in, const int* in_sizes, int n_in,
                              void* d_out, int out_size, void* d_ws, size_t ws_size,
                              hipStream_t stream) {
    const float* x = (const float*)d_in[0];
    const float* cores[7];
    for (int d = 0; d < 7; ++d) cores[d] = (const float*)d_in[1 + d];

    float* buf0 = (float*)d_ws;
    float* buf1 = buf0 + (size_t)262144 * 64;   // 67 MB each, ping-pong

    dim3 block(256);
    tt_first<<<dim3(2048), block, 0, stream>>>(x, cores[6], buf0);
    tt_mid<<<dim3(1024), block, 0, stream>>>(buf0, cores[5], buf1, 3);
    tt_mid<<<dim3(1024), block, 0, stream>>>(buf1, cores[4], buf0, 6);
    tt_mid<<<dim3(1024), block, 0, stream>>>(buf0, cores[3], buf1, 9);
    tt_mid<<<dim3(1024), block, 0, stream>>>(buf1, cores[2], buf0, 12);
    tt_mid<<<dim3(1024), block, 0, stream>>>(buf0, cores[1], buf1, 15);
    tt_last<<<dim3(2048), block, 0, stream>>>(buf1, cores[0], (float*)d_out);
}